// Chamfer_Loss_18262200943427
// MI455X (gfx1250) — compile-verified
//
#include <hip/hip_runtime.h>

typedef float v2f __attribute__((ext_vector_type(2)));
typedef float v8f __attribute__((ext_vector_type(8)));

#define INF_BITS 0x7F800000u
#define NT      4   // 16-row n-tiles per wave (4 WMMAs share each B fragment)
#define MCHUNKS 8   // waves splitting each row-group's M range

// ds_swizzle xor-butterfly (group-of-32 mode: offset = xor<<10 | and_mask 0x1f)
#define SWZU(x, imm) ((unsigned)__builtin_amdgcn_ds_swizzle((int)(x), (imm)))

static __device__ __forceinline__ unsigned uminu(unsigned a, unsigned b) {
    return a < b ? a : b;   // v_min_u32 — no NaN canonicalization needed
}

__global__ void init_mins(unsigned* __restrict__ p, int n) {
    int i = blockIdx.x * blockDim.x + threadIdx.x;
    if (i < n) p[i] = INF_BITS;
}

// One wave = (batch b, 64-row n-group, M/MCHUNKS column chunk).
// Per m-tile: one B fragment feeds NT=4 V_WMMA_F32_16X16X4_F32 ops via the
// augmented rank-4 trick:
//   A row i = [-2ax, -2ay, -2az, |a|^2]   (16x4 f32)
//   B col j = [ bx,   by,   bz,  1   ]    (4x16 f32)
//   D = A*B (+ inline C=0), then d2 = max(D + |b_j|^2, 0)
// After the clamp all values are finite and >= 0, so uint bit patterns are
// order-preserving: every min (register accumulators, ds_swizzle butterfly,
// global atomics) runs as integer min with zero canonicalization overhead.
// The pred-point load is software-pipelined one m-tile ahead so the
// s_wait_loadcnt lands a full compute body after issue.
__global__ __launch_bounds__(256) void chamfer_tiles(
    const float* __restrict__ inputs,   // [B,N,3]
    const float* __restrict__ preds,    // [B,M,3]
    unsigned* __restrict__ min1u,       // [B*M] min over n
    unsigned* __restrict__ min2u,       // [B*N] min over m
    int N, int M)
{
    const int lane = threadIdx.x & 31;
    const int wid  = (blockIdx.x * blockDim.x + threadIdx.x) >> 5;
    const int ngroups = N >> 6;                    // 64 rows per group
    const int tiles_per_chunk = (M >> 4) / MCHUNKS;

    const int mc = wid % MCHUNKS;
    const int ng = (wid / MCHUNKS) % ngroups;
    const int b  = wid / (MCHUNKS * ngroups);

    const int n0 = ng << 6;
    const int l  = lane & 15;   // row/col index within a 16-wide tile
    const int hi = lane >> 4;   // lane half selects K pair (ISA 16x4 f32 layout)
    const int l3 = l * 3;

    // ---- NT A fragments + u32 row-min accumulators (persistent) ----
    const float* ap = inputs + ((size_t)b * N + n0) * 3;
    v2f afrag[NT];
    unsigned rowAcc[NT][8];
#pragma unroll
    for (int j = 0; j < NT; ++j) {
        const float* pj = ap + j * 48 + l3;
        float px = pj[0], py = pj[1], pz = pj[2];
        float n2 = px * px + py * py + pz * pz;
        afrag[j].x = hi ? (-2.0f * pz) : (-2.0f * px);
        afrag[j].y = hi ? n2           : (-2.0f * py);
#pragma unroll
        for (int i = 0; i < 8; ++i) rowAcc[j][i] = INF_BITS;
    }

    const float* qbase = preds + (size_t)b * M * 3 + (size_t)mc * tiles_per_chunk * 48;
    unsigned* col_out  = min1u + (size_t)b * M + (size_t)mc * tiles_per_chunk * 16;

    const v8f czero = {};   // SRC2 inline-constant 0 in the WMMA encoding

    // prologue: fetch tile 0's pred point
    float qx = qbase[l3 + 0], qy = qbase[l3 + 1], qz = qbase[l3 + 2];

#pragma unroll 2
    for (int t = 0; t < tiles_per_chunk; ++t) {
        // ---- issue next tile's load now; consume it next iteration ----
        const int tn = (t + 1 < tiles_per_chunk) ? t + 1 : t;   // s_min, no OOB
        const float* qn = qbase + tn * 48 + l3;
        float nx = qn[0], ny = qn[1], nz = qn[2];
        __builtin_prefetch(qbase + (t + 2) * 48 + l3, 0, 3);    // L0 warm, 2 ahead

        float m2 = qx * qx + qy * qy + qz * qz;
        v2f bfrag;
        bfrag.x = hi ? qz   : qx;
        bfrag.y = hi ? 1.0f : qy;

        unsigned cm = INF_BITS;     // this half's column min over NT*8 rows
#pragma unroll
        for (int j = 0; j < NT; ++j) {
            v8f d = __builtin_amdgcn_wmma_f32_16x16x4_f32(
                /*neg_a=*/false, afrag[j], /*neg_b=*/false, bfrag,
                /*c_mod=*/(short)0, czero, /*reuse_a=*/false, /*reuse_b=*/false);
#pragma unroll
            for (int i = 0; i < 8; ++i) {
                unsigned e = __float_as_uint(fmaxf(d[i] + m2, 0.0f));
                rowAcc[j][i] = uminu(rowAcc[j][i], e);
                cm = uminu(cm, e);
            }
        }
        // Both lane halves min into the same column slot: no shuffle, no
        // divergence -> EXEC stays all-ones across every WMMA in the loop.
        atomicMin(&col_out[t * 16 + l], cm);

        qx = nx; qy = ny; qz = nz;   // rotate pipeline registers
    }

    // ---- tail: u32 xor-butterfly across each 16-lane half ----
    unsigned* row_out = min2u + (size_t)b * N + n0;
#pragma unroll
    for (int j = 0; j < NT; ++j) {
#pragma unroll
        for (int r = 0; r < 8; ++r) {
            unsigned v = rowAcc[j][r];
            v = uminu(v, SWZU(v, 0x041f));   // xor 1
            v = uminu(v, SWZU(v, 0x081f));   // xor 2
            v = uminu(v, SWZU(v, 0x101f));   // xor 4
            v = uminu(v, SWZU(v, 0x201f));   // xor 8
            // lane 0 -> row n0+j*16+r ; lane 16 -> row n0+j*16+r+8
            if (l == 0)
                atomicMin(&row_out[j * 16 + r + hi * 8], v);
        }
    }
}

__global__ __launch_bounds__(256) void finalize_loss(
    const unsigned* __restrict__ min1u, const unsigned* __restrict__ min2u,
    int c1, int c2, float* __restrict__ out)
{
    __shared__ float s_sum[256];
    __shared__ float s_max[256];
    const int tid = threadIdx.x;
    float loss[2];
    const unsigned* ptrs[2] = { min1u, min2u };
    const int cnts[2] = { c1, c2 };

    for (int k = 0; k < 2; ++k) {
        float sum = 0.0f, mx = -__builtin_inff();
        for (int i = tid; i < cnts[k]; i += 256) {
            float v = __uint_as_float(ptrs[k][i]);   // values >= 0: bitcast exact
            sum += v;
            mx = fmaxf(mx, v);
        }
        s_sum[tid] = sum;
        s_max[tid] = mx;
        __syncthreads();
        for (int s = 128; s > 0; s >>= 1) {
            if (tid < s) {
                s_sum[tid] += s_sum[tid + s];
                s_max[tid]  = fmaxf(s_max[tid], s_max[tid + s]);
            }
            __syncthreads();
        }
        loss[k] = s_sum[0] / (float)cnts[k] + s_max[0];
        __syncthreads();
    }
    if (tid == 0) out[0] = fmaxf(loss[0], loss[1]);
}

extern "C" void kernel_launch(void* const* d_in, const int* in_sizes, int n_in,
                              void* d_out, int out_size, void* d_ws, size_t ws_size,
                              hipStream_t stream) {
    const float* inputs = (const float*)d_in[0];
    const float* preds  = (const float*)d_in[1];
    const int B = 4, D = 3;
    const int N = in_sizes[0] / (B * D);   // 4096
    const int M = in_sizes[1] / (B * D);   // 4096

    unsigned* min1u = (unsigned*)d_ws;            // B*M
    unsigned* min2u = min1u + (size_t)B * M;      // B*N
    const int tot = B * (M + N);
    init_mins<<<(tot + 255) / 256, 256, 0, stream>>>(min1u, tot);

    const int waves = B * (N / (16 * NT)) * MCHUNKS;   // 2048 waves, fully active
    chamfer_tiles<<<(waves * 32) / 256, 256, 0, stream>>>(
        inputs, preds, min1u, min2u, N, M);

    finalize_loss<<<1, 256, 0, stream>>>(min1u, min2u, B * M, B * N, (float*)d_out);
}